// CustomModel_1735166788001
// MI455X (gfx1250) — compile-verified
//
#include <hip/hip_runtime.h>

// ---------------- problem constants ----------------
#define NN   86016      // nodes
#define EE   4194304    // directed edges
#define ET   4280320    // EE + NN (with self loops)
#define HH   32         // hidden
#define BB   1024       // graphs
#define NPG  84         // nodes per graph
#define NEG_SLOPE 0.2f

typedef float v2f __attribute__((ext_vector_type(2)));
typedef float v8f __attribute__((ext_vector_type(8)));

// gfx1250 async global->LDS copy support (ASYNCcnt-tracked).
// Probe-confirmed param types: (int32x4 AS1*, int32x4 AS3*, imm int, imm int)
#if defined(__has_builtin)
#  if __has_builtin(__builtin_amdgcn_global_load_async_to_lds_b128)
#    define HAVE_ASYNC_LDS 1
#  else
#    define HAVE_ASYNC_LDS 0
#  endif
#else
#  define HAVE_ASYNC_LDS 0
#endif

typedef int b128_t __attribute__((__vector_size__(4 * sizeof(int))));
typedef __attribute__((address_space(1))) b128_t glob_b128;
typedef __attribute__((address_space(3))) b128_t lds_b128;

__device__ __forceinline__ void wait_async_zero() {
#if defined(__has_builtin)
#  if __has_builtin(__builtin_amdgcn_s_wait_asynccnt)
    __builtin_amdgcn_s_wait_asynccnt(0);
    return;
#  endif
#endif
    asm volatile("s_wait_asynccnt 0x0" ::: "memory");
}

// ---------------- small utility kernels ----------------
__global__ void fill_f32_kernel(float* __restrict__ p, float v, long n) {
    long i = (long)blockIdx.x * blockDim.x + threadIdx.x;
    if (i < n) p[i] = v;
}
__global__ void fill_u32_kernel(unsigned* __restrict__ p, unsigned v, long n) {
    long i = (long)blockIdx.x * blockDim.x + threadIdx.x;
    if (i < n) p[i] = v;
}
// out[node][f] = bias[f]  (accumulator init; atomics add on top)
__global__ void bias_init_kernel(float* __restrict__ hA, const float* __restrict__ bias_l) {
    long i = (long)blockIdx.x * blockDim.x + threadIdx.x;
    if (i < (long)NN * HH) hA[i] = bias_l[i & (HH - 1)];
}

// ---------------- self-loop attr: mean of incoming edge attrs ----------------
__global__ void deg_easum_kernel(const int* __restrict__ dst, const float* __restrict__ ea,
                                 float* __restrict__ deg, float* __restrict__ easum) {
    long e = (long)blockIdx.x * blockDim.x + threadIdx.x;
    if (e >= EE) return;
    int d = dst[e];
    atomicAdd(&deg[d], 1.0f);
    atomicAdd(&easum[(long)d * 2 + 0], ea[e * 2 + 0]);
    atomicAdd(&easum[(long)d * 2 + 1], ea[e * 2 + 1]);
}
__global__ void ealoop_kernel(const float* __restrict__ deg, const float* __restrict__ easum,
                              float* __restrict__ ealoop) {
    long i = (long)blockIdx.x * blockDim.x + threadIdx.x;
    if (i >= (long)NN * 2) return;
    float dg = fmaxf(deg[i >> 1], 1.0f);
    ealoop[i] = easum[i] / dg;
}

// c01 = We[l] @ att_edge[l]  ([2,32]@[32] -> 2 floats); 1 block of 64 threads
__global__ void c01_kernel(const float* __restrict__ We_l, const float* __restrict__ ae_l,
                           float* __restrict__ c01) {
    int lane = threadIdx.x & 31;
    int j    = threadIdx.x >> 5;   // 0 or 1
    float p = We_l[j * HH + lane] * ae_l[lane];
    for (int off = 16; off >= 1; off >>= 1) p += __shfl_xor(p, off);
    if (lane == 0) c01[j] = p;
}

// ---------------- layer 0 transform: h = x @ W0 ([N,1]@[1,32]) + alphas ----------------
__global__ __launch_bounds__(256)
void layer0_transform_kernel(const float* __restrict__ x, const float* __restrict__ W0,
                             const float* __restrict__ asw, const float* __restrict__ adw,
                             float* __restrict__ hB, float* __restrict__ asrc, float* __restrict__ adst) {
    long idx = (long)blockIdx.x * blockDim.x + threadIdx.x;
    if (idx >= (long)NN * HH) return;
    int node = (int)(idx >> 5);
    int n    = (int)(idx & 31);         // lane == feature
    float h = x[node] * W0[n];
    hB[idx] = h;
    float va = h * asw[n];
    float vd = h * adw[n];
    for (int off = 16; off >= 1; off >>= 1) {
        va += __shfl_xor(va, off);
        vd += __shfl_xor(vd, off);
    }
    if (n == 0) { asrc[node] = va; adst[node] = vd; }
}

// ---------------- layers 1..3 transform via fp32 WMMA -------------------------
// h_t = relu(h_in) @ W   (relu of previous layer folded into the LDS read)
// Block tile = 128 rows staged into LDS via GLOBAL_LOAD_ASYNC_TO_LDS_B128
// (coalesced 128B bursts, ASYNCcnt), then 8 waves each run a 16-row tile:
// 8 K-steps of V_WMMA_F32_16X16X4_F32 per 16-col tile (16 WMMAs/wave).
// alpha_src/alpha_dst are reduced straight from the C fragments.
__global__ __launch_bounds__(256)
void wmma_transform_kernel(const float* __restrict__ hin, const float* __restrict__ W,
                           const float* __restrict__ as_w, const float* __restrict__ ad_w,
                           float* __restrict__ hout, float* __restrict__ asrc, float* __restrict__ adst) {
    __shared__ float Hl[128 * HH];   // 16 KB block tile of input rows
    __shared__ float Wl[HH * HH];    // 4 KB weights
    __shared__ float Asv[HH], Adv[HH];
    int tid = threadIdx.x;
    const float* gH = hin + (size_t)blockIdx.x * 128 * HH;

#if HAVE_ASYNC_LDS
    {
        int i = tid * 4;                       // 16B per thread per shot
        #pragma unroll
        for (int it = 0; it < 4; ++it, i += 1024)
            __builtin_amdgcn_global_load_async_to_lds_b128(
                (glob_b128*)(gH + i), (lds_b128*)(Hl + i), 0, 0);
        __builtin_amdgcn_global_load_async_to_lds_b128(
            (glob_b128*)(W + tid * 4), (lds_b128*)(Wl + tid * 4), 0, 0);
    }
    if (tid < HH) { Asv[tid] = as_w[tid]; Adv[tid] = ad_w[tid]; }
    wait_async_zero();
    __syncthreads();
#else
    for (int i = tid; i < 128 * HH; i += 256) Hl[i] = gH[i];
    for (int i = tid; i < HH * HH; i += 256)  Wl[i] = W[i];
    if (tid < HH) { Asv[tid] = as_w[tid]; Adv[tid] = ad_w[tid]; }
    __syncthreads();
#endif

    int wave = tid >> 5, lane = tid & 31;
    int rb   = (blockIdx.x * 8 + wave) * 16;   // global row base of this wave's tile
    int m    = lane & 15;                      // A-row / B-col within tile
    int g    = lane >> 4;                      // K half selector (ISA 16x4 f32 layout)
    int n0 = m, n1 = m + 16;

    v8f c0 = {}; v8f c1 = {};
    const float* hrow = Hl + (wave * 16 + m) * HH;
    #pragma unroll
    for (int kk = 0; kk < 8; ++kk) {
        int klo = kk * 4 + g * 2;
        v2f a;  a.x  = fmaxf(hrow[klo], 0.0f);      a.y  = fmaxf(hrow[klo + 1], 0.0f);
        v2f b0; b0.x = Wl[klo * HH + n0];           b0.y = Wl[(klo + 1) * HH + n0];
        v2f b1; b1.x = Wl[klo * HH + n1];           b1.y = Wl[(klo + 1) * HH + n1];
        c0 = __builtin_amdgcn_wmma_f32_16x16x4_f32(false, a, false, b0, (short)0, c0, false, false);
        c1 = __builtin_amdgcn_wmma_f32_16x16x4_f32(false, a, false, b1, (short)0, c1, false, false);
    }

    float sA0 = Asv[n0], sA1 = Asv[n1];
    float sD0 = Adv[n0], sD1 = Adv[n1];
    #pragma unroll
    for (int r = 0; r < 8; ++r) {
        int orow = rb + g * 8 + r;          // C layout: VGPR r -> row r (lanes 0-15), r+8 (16-31)
        hout[(long)orow * HH + n0] = c0[r];
        hout[(long)orow * HH + n1] = c1[r];
        float va = c0[r] * sA0 + c1[r] * sA1;
        float vd = c0[r] * sD0 + c1[r] * sD1;
        for (int off = 8; off >= 1; off >>= 1) {   // reduce across the 16 lanes of each half
            va += __shfl_xor(va, off);
            vd += __shfl_xor(vd, off);
        }
        if (m == 0) { asrc[orow] = va; adst[orow] = vd; }
    }
}

// ---------------- edge pass 1: logits + segment max (monotone-uint atomicMax) ----
__device__ __forceinline__ unsigned f32_order_key(float f) {
    unsigned u = __float_as_uint(f);
    return (u & 0x80000000u) ? ~u : (u | 0x80000000u);
}
__device__ __forceinline__ float f32_order_unkey(unsigned k) {
    unsigned u = (k & 0x80000000u) ? (k & 0x7fffffffu) : ~k;
    return __uint_as_float(u);
}
__global__ __launch_bounds__(256)
void logits_max_kernel(const int* __restrict__ src, const int* __restrict__ dst,
                       const float* __restrict__ ea, const float* __restrict__ ealoop,
                       const float* __restrict__ asv, const float* __restrict__ adv,
                       const float* __restrict__ c01,
                       float* __restrict__ logw, unsigned* __restrict__ maxb) {
    long e = (long)blockIdx.x * blockDim.x + threadIdx.x;
    if (e >= ET) return;
    int s, d; float a0, a1;
    if (e < EE) {
        s = src[e]; d = dst[e];
        a0 = ea[e * 2 + 0]; a1 = ea[e * 2 + 1];
    } else {
        s = d = (int)(e - EE);
        a0 = ealoop[(long)s * 2 + 0]; a1 = ealoop[(long)s * 2 + 1];
    }
    float lg = asv[s] + adv[d] + a0 * c01[0] + a1 * c01[1];
    lg = (lg >= 0.0f) ? lg : NEG_SLOPE * lg;        // leaky_relu(0.2)
    logw[e] = lg;
    atomicMax(&maxb[d], f32_order_key(lg));
}

// ---------------- edge pass 2: w = exp(logit - max[dst]); denom += w ------------
__global__ __launch_bounds__(256)
void exp_denom_kernel(const int* __restrict__ dst, float* __restrict__ logw,
                      const unsigned* __restrict__ maxb, float* __restrict__ denom) {
    long e = (long)blockIdx.x * blockDim.x + threadIdx.x;
    if (e >= ET) return;
    int d = (e < EE) ? dst[e] : (int)(e - EE);
    float m = f32_order_unkey(maxb[d]);
    float w = __expf(logw[e] - m);
    logw[e] = w;
    atomicAdd(&denom[d], w);
}

// ---------------- edge pass 3: out[dst] += (w/denom[dst]) * h[src] --------------
// wave32: lane == feature. Each wave loads 32 edges' meta coalesced, then
// broadcasts (src,dst,alpha) and does 128B coalesced gather + 32 f32 atomics.
__global__ __launch_bounds__(256)
void aggregate_kernel(const int* __restrict__ src, const int* __restrict__ dst,
                      const float* __restrict__ w, const float* __restrict__ denom,
                      const float* __restrict__ hB, float* __restrict__ out) {
    int lane = threadIdx.x & 31;
    long wid  = (long)blockIdx.x * 8 + (threadIdx.x >> 5);
    long base = wid * 32;
    if (base >= ET) return;
    int cnt = (int)((ET - base) < 32 ? (ET - base) : 32);

    int s = 0, d = 0; float alpha = 0.0f;
    long e = base + lane;
    if (lane < cnt) {
        if (e < EE) { s = src[e]; d = dst[e]; }
        else        { s = d = (int)(e - EE); }
        alpha = w[e] / denom[d];
    }
    for (int j = 0; j < cnt; ++j) {
        int   ss = __shfl(s, j);
        int   dd = __shfl(d, j);
        float aa = __shfl(alpha, j);
        float v  = aa * hB[(long)ss * HH + lane];
        atomicAdd(&out[(long)dd * HH + lane], v);
    }
}

// ---------------- pooling: sum 84 nodes, dot lin_w, +lin_b, relu ---------------
__global__ __launch_bounds__(256)
void pool_kernel(const float* __restrict__ h, const float* __restrict__ lw,
                 const float* __restrict__ lb, float* __restrict__ out) {
    int lane = threadIdx.x & 31;
    int g = blockIdx.x * 8 + (threadIdx.x >> 5);
    if (g >= BB) return;
    const float* base = h + (long)g * NPG * HH;
    float acc = 0.0f;
    #pragma unroll 4
    for (int i = 0; i < NPG; ++i) acc += base[i * HH + lane];
    float p = acc * lw[lane];
    for (int off = 16; off >= 1; off >>= 1) p += __shfl_xor(p, off);
    if (lane == 0) out[g] = fmaxf(p + lb[0], 0.0f);
}

// ---------------- host orchestration ----------------
extern "C" void kernel_launch(void* const* d_in, const int* in_sizes, int n_in,
                              void* d_out, int out_size, void* d_ws, size_t ws_size,
                              hipStream_t stream) {
    const float* x        = (const float*)d_in[0];
    const int*   ei       = (const int*)d_in[1];
    const float* ea       = (const float*)d_in[2];
    const float* W0       = (const float*)d_in[3];
    const float* Ws       = (const float*)d_in[4];   // [3,32,32]
    const float* att_src  = (const float*)d_in[5];   // [4,32]
    const float* att_dst  = (const float*)d_in[6];   // [4,32]
    const float* We       = (const float*)d_in[7];   // [4,2,32]
    const float* att_edge = (const float*)d_in[8];   // [4,32]
    const float* bias     = (const float*)d_in[9];   // [4,32]
    const float* lin_w    = (const float*)d_in[10];  // [32]
    const float* lin_b    = (const float*)d_in[11];  // [1]
    float* out = (float*)d_out;

    const int* src  = ei;
    const int* dstp = ei + EE;

    // ---- workspace carve-out (~42 MB) ----
    char* p = (char*)d_ws;
    auto carve = [&](size_t bytes) { char* r = p; p += (bytes + 255) & ~(size_t)255; return r; };
    float*    hA     = (float*)   carve((size_t)NN * HH * 4);  // aggregate / layer in-out
    float*    hB     = (float*)   carve((size_t)NN * HH * 4);  // transformed features
    float*    asrc   = (float*)   carve((size_t)NN * 4);
    float*    adst   = (float*)   carve((size_t)NN * 4);
    float*    ealoop = (float*)   carve((size_t)NN * 2 * 4);
    float*    deg    = (float*)   carve((size_t)NN * 4);
    float*    easum  = (float*)   carve((size_t)NN * 2 * 4);
    float*    logw   = (float*)   carve((size_t)ET * 4);
    unsigned* maxb   = (unsigned*)carve((size_t)NN * 4);
    float*    denom  = (float*)   carve((size_t)NN * 4);
    float*    c01    = (float*)   carve(256);

    const int T = 256;
    dim3 blk(T);
    const long NH      = (long)NN * HH;
    const int  gNH     = (int)((NH + T - 1) / T);       // 10752
    const int  gN      = (NN + T - 1) / T;              // 336
    const int  gN2     = (2 * NN + T - 1) / T;          // 672
    const int  gE      = (EE + T - 1) / T;              // 16384
    const int  gET     = (int)(((long)ET + T - 1) / T); // 16720
    const int  gAgg    = (int)(((long)ET / 32 + 7) / 8);// waves of 32 edges, 8 waves/block
    const int  gWmma   = NN / 128;                      // 672 (8 tiles of 16 rows / block)

    // ---- self-loop edge-attr means (fill_value='mean') ----
    fill_f32_kernel<<<gN,  blk, 0, stream>>>(deg,   0.0f, NN);
    fill_f32_kernel<<<gN2, blk, 0, stream>>>(easum, 0.0f, (long)NN * 2);
    deg_easum_kernel<<<gE, blk, 0, stream>>>(dstp, ea, deg, easum);
    ealoop_kernel<<<gN2, blk, 0, stream>>>(deg, easum, ealoop);

    // ---- 4 GAT layers ----
    for (int l = 0; l < 4; ++l) {
        c01_kernel<<<1, 64, 0, stream>>>(We + l * 64, att_edge + l * HH, c01);

        if (l == 0) {
            layer0_transform_kernel<<<gNH, blk, 0, stream>>>(
                x, W0, att_src, att_dst, hB, asrc, adst);
        } else {
            wmma_transform_kernel<<<gWmma, blk, 0, stream>>>(
                hA, Ws + (size_t)(l - 1) * HH * HH,
                att_src + l * HH, att_dst + l * HH, hB, asrc, adst);
        }

        fill_u32_kernel<<<gN, blk, 0, stream>>>(maxb, 0u, NN);     // encoded -inf
        fill_f32_kernel<<<gN, blk, 0, stream>>>(denom, 0.0f, NN);
        bias_init_kernel<<<gNH, blk, 0, stream>>>(hA, bias + l * HH);

        logits_max_kernel<<<gET, blk, 0, stream>>>(src, dstp, ea, ealoop,
                                                   asrc, adst, c01, logw, maxb);
        exp_denom_kernel<<<gET, blk, 0, stream>>>(dstp, logw, maxb, denom);
        aggregate_kernel<<<gAgg, blk, 0, stream>>>(src, dstp, logw, denom, hB, hA);
        // relu between layers is folded into the next wmma_transform's input load;
        // final layer output (hA) is consumed raw by pooling, as in the reference.
    }

    // ---- pool + linear + relu ----
    pool_kernel<<<BB / 8, blk, 0, stream>>>(hA, lin_w, lin_b, out);
}